// FratPiror_25812753448989
// MI455X (gfx1250) — compile-verified
//
#include <hip/hip_runtime.h>
#include <math.h>

typedef __attribute__((ext_vector_type(2))) float v2f;
typedef __attribute__((ext_vector_type(8))) float v8f;

#define IMGSZ 224
#define IMG2  (IMGSZ * IMGSZ)   // 50176
#define NPIX  (4 * IMG2)        // 200704 pixel positions (B * H * W)

__global__ __launch_bounds__(256)
void frat_fused_kernel(const float* __restrict__ x,
                       const float* __restrict__ Wc,
                       const float* __restrict__ bias,
                       float* __restrict__ out)
{
    const int pix  = blockIdx.x * 256 + threadIdx.x;   // 0 .. NPIX-1, exact cover
    const int lane = threadIdx.x & 31;

    const int bimg = pix / IMG2;
    const int sp   = pix - bimg * IMG2;
    const int i    = sp / IMGSZ;
    const int j    = sp - i * IMGSZ;

    float yv[3];

    #pragma unroll
    for (int c = 0; c < 3; ++c) {
        const float* __restrict__ xc = x + (bimg * 3 + c) * IMG2;

        // 7x7 zero-padded patch, fully register-resident after unroll
        float P[49];
        #pragma unroll
        for (int ki = 0; ki < 7; ++ki) {
            const int ii  = i + ki - 3;
            const bool rk = (ii >= 0) && (ii < IMGSZ);
            #pragma unroll
            for (int kj = 0; kj < 7; ++kj) {
                const int jj = j + kj - 3;
                const bool ok = rk && (jj >= 0) && (jj < IMGSZ);
                P[ki * 7 + kj] = ok ? xc[ii * IMGSZ + jj] : 0.0f;
            }
        }

        // ---- s = 2, m = 6 : 36 groups of 4 consecutive raw indices ----
        float tp2 = 0.0f;
        #pragma unroll
        for (int p = 0; p < 36; ++p) {
            float mx = -__builtin_inff(), mn = __builtin_inff();
            #pragma unroll
            for (int t = 0; t < 4; ++t) {
                const int u   = p * 4 + t;
                const int q   = u / 36;            // si*2 + sj
                const int rem = u - q * 36;
                const int si  = q >> 1, sj = q & 1;
                const int mi  = rem / 6, mj = rem - mi * 6;
                const float v = P[(mi + si) * 7 + (mj + sj)];
                mx = fmaxf(mx, v); mn = fminf(mn, v);
            }
            tp2 += (mx - mn) * 3.5f + 1.0f;        // rng / (2/7) + 1
        }

        // ---- s = 3, m = 5 : 25 groups of 9 (groups straddle 25-blocks) ----
        float tp3 = 0.0f;
        #pragma unroll
        for (int p = 0; p < 25; ++p) {
            float mx = -__builtin_inff(), mn = __builtin_inff();
            #pragma unroll
            for (int t = 0; t < 9; ++t) {
                const int u   = p * 9 + t;
                const int q   = u / 25;            // si*3 + sj
                const int rem = u - q * 25;
                const int si  = q / 3,  sj = q - si * 3;
                const int mi  = rem / 5, mj = rem - mi * 5;
                const float v = P[(mi + si) * 7 + (mj + sj)];
                mx = fmaxf(mx, v); mn = fminf(mn, v);
            }
            tp3 += (mx - mn) * (7.0f / 3.0f) + 1.0f;   // rng / (3/7) + 1
        }

        const float ws = 0.5f * (logf(tp2) + logf(tp3));
        yv[c] = ws + P[24];                        // P[24] == x[b,c,i,j]
    }

    const float y0 = yv[0], y1 = yv[1], y2 = yv[2];

    // ================= 1x1 conv via V_WMMA_F32_16X16X4_F32 =================
    // D[M][N] = sum_K A[M][K]*B[K][N] + C[M][N]
    //   M = out channel (rows 0..2 used of 16), K = in channel (0..2 of 4),
    //   N = pixel (16 per WMMA; two WMMAs cover the wave's 32 pixels).
    // f32 A/B layout: 2 VGPRs; lanes 0-15 hold K=0 (v0) / K=1 (v1),
    //                 lanes 16-31 hold K=2 (v0) / K=3 (v1).
    const int r = lane & 15;

    float a0 = 0.0f, a1 = 0.0f;
    if (r < 3) {
        if (lane < 16) { a0 = Wc[r * 3 + 0]; a1 = Wc[r * 3 + 1]; }
        else           { a0 = Wc[r * 3 + 2]; }
    }
    v2f A; A.x = a0; A.y = a1;

    // C accumulator carries the bias: VGPR v, lanes 0-15 -> M=v (M=0..2 used)
    v8f Cm = {};
    if (lane < 16) { Cm[0] = bias[0]; Cm[1] = bias[1]; Cm[2] = bias[2]; }

    // B operand packing (uniform shuffles, then select):
    const float t2 = __shfl(y2, r,      32);   // y2 of lane r       (for lanes 16-31, WMMA#1)
    const float t0 = __shfl(y0, r + 16, 32);   // y0 of lane r+16    (for lanes 0-15,  WMMA#2)
    const float t1 = __shfl(y1, r + 16, 32);   // y1 of lane r+16

    v2f B1, B2;
    B1.x = (lane < 16) ? y0 : t2;   // K=0 rows | K=2 rows, pixels base+0..15
    B1.y = (lane < 16) ? y1 : 0.0f; // K=1 rows | K=3 pad
    B2.x = (lane < 16) ? t0 : y2;   // pixels base+16..31
    B2.y = (lane < 16) ? t1 : 0.0f;

    // EXEC is all-ones here (grid exactly covers NPIX, no early exits).
    v8f D1 = __builtin_amdgcn_wmma_f32_16x16x4_f32(false, A, false, B1,
                                                   (short)0, Cm, false, false);
    v8f D2 = __builtin_amdgcn_wmma_f32_16x16x4_f32(false, A, false, B2,
                                                   (short)0, Cm, false, false);

    // D layout: VGPR o, lanes 0-15 -> out channel o, pixel N=lane.
    if (lane < 16) {
        const int wbase = pix - lane;         // wave's base pixel
        const int p1 = wbase + lane;          // == pix
        const int p2 = wbase + 16 + lane;
        const int b1 = p1 / IMG2, s1 = p1 - b1 * IMG2;
        const int b2 = p2 / IMG2, s2 = p2 - b2 * IMG2;
        #pragma unroll
        for (int o = 0; o < 3; ++o) {
            out[(b1 * 3 + o) * IMG2 + s1] = D1[o];
            out[(b2 * 3 + o) * IMG2 + s2] = D2[o];
        }
    }
}

extern "C" void kernel_launch(void* const* d_in, const int* in_sizes, int n_in,
                              void* d_out, int out_size, void* d_ws, size_t ws_size,
                              hipStream_t stream) {
    const float* x  = (const float*)d_in[0];   // [4,3,224,224]
    const float* Wc = (const float*)d_in[1];   // [3,3] row-major [o][c]
    const float* b  = (const float*)d_in[2];   // [3]
    float* out = (float*)d_out;                // [4,3,224,224]

    const int threads = 256;
    const int blocks  = NPIX / threads;        // 784, exact
    frat_fused_kernel<<<blocks, threads, 0, stream>>>(x, Wc, b, out);
}